// WeightFn_68822555951213
// MI455X (gfx1250) — compile-verified
//
#include <hip/hip_runtime.h>
#include <hip/hip_bf16.h>
#include <math.h>

typedef __attribute__((ext_vector_type(2))) float v2f;
typedef __attribute__((ext_vector_type(4))) float v4f;
typedef __attribute__((ext_vector_type(8))) float v8f;

#define BB 256
#define MM 8192
#define WW 32
#define HH 4
#define EPSV 1e-5f
#define SSTR 8208            // padded LDS stride (floats); 8208%64==16 -> store lanes hit distinct banks
#define NTHREADS 1024
#define NWAVES 32
#define NTILES (MM / 16)     // 512 tiles of 16 memory rows

__device__ __forceinline__ float wave_max(float v) {
#pragma unroll
    for (int o = 16; o > 0; o >>= 1) v = fmaxf(v, __shfl_xor(v, o, 32));
    return v;
}
__device__ __forceinline__ float wave_sum(float v) {
#pragma unroll
    for (int o = 16; o > 0; o >>= 1) v += __shfl_xor(v, o, 32);
    return v;
}

__global__ __launch_bounds__(NTHREADS, 1)
void cosine_addr_softmax_kernel(const float* __restrict__ memory,
                                const float* __restrict__ keys,
                                const float* __restrict__ strengths,
                                float* __restrict__ out)
{
    extern __shared__ float smem[];
    float* dotLds   = smem;                       // [4][SSTR] raw dot products
    float* mnormLds = smem + 4 * SSTR;            // [MM] memory row norms
    float* red      = smem + 4 * SSTR + MM;       // [64] reduction / broadcast scratch

    const int b    = blockIdx.x;
    const int tid  = threadIdx.x;
    const int lane = tid & 31;
    const int wid  = tid >> 5;
    const int row  = lane & 15;   // A-matrix row within 16-row tile
    const int kh   = lane >> 4;   // K-half (A VGPR0 holds K=0/K=2 halves per ISA layout)
    const int hval = lane & 15;   // D column (head index) owned by this lane
    const int hc   = hval & 3;    // clamped head for safe addressing

    const float* memb = memory + (size_t)b * MM * WW;
    const float* keyb = keys   + (size_t)b * HH * WW;

    // ---- per-lane key-side constants (key norm, softplus(strength)) ----
    float ks = 0.f;
#pragma unroll
    for (int w = 0; w < WW; ++w) { float kv = keyb[hc * WW + w]; ks += kv * kv; }
    const float keynorm = __builtin_amdgcn_sqrtf(ks + EPSV);
    const float st = strengths[(size_t)b * HH + hc];
    const float sp = (st > 20.f) ? st : log1pf(__expf(st));     // softplus
    if (tid < HH) { red[48 + tid] = keynorm; red[52 + tid] = sp; }   // broadcast for phase 2

    // ---- B fragments: keys^T per K-chunk (4x16), heads 4..15 zero-padded ----
    const float hmask = (hval < HH) ? 1.f : 0.f;
    v2f bf[8];
#pragma unroll
    for (int w = 0; w < 8; ++w) {
        bf[w].x = hmask * keyb[hc * WW + w * 4 + kh * 2 + 0];
        bf[w].y = hmask * keyb[hc * WW + w * 4 + kh * 2 + 1];
    }

    // ---- phase 1: stream memory once; WMMA dots + row norms -> LDS (raw) ----
    for (int t = wid; t < NTILES; t += NWAVES) {
        const int mbase = t * 16;
        const float* rowp = memb + (size_t)(mbase + row) * WW + kh * 2;
        if (t + NWAVES < NTILES)
            __builtin_prefetch(memb + (size_t)(mbase + NWAVES * 16 + row) * WW + kh * 2, 0, 1);

        v8f c = {0.f, 0.f, 0.f, 0.f, 0.f, 0.f, 0.f, 0.f};
        float ss = 0.f;
#pragma unroll
        for (int w = 0; w < 8; ++w) {
            v2f a = *(const v2f*)(rowp + w * 4);
            ss += a.x * a.x + a.y * a.y;
            // D = A(16x4 f32) * B(4x16 f32) + C, full f32 precision
            c = __builtin_amdgcn_wmma_f32_16x16x4_f32(false, a, false, bf[w],
                                                      (short)0, c, false, false);
        }
        // row sum-of-squares: combine the two K-halves (lane r <-> r+16)
        ss += __shfl_xor(ss, 16, 32);
        const float rnorm = __builtin_amdgcn_sqrtf(ss + EPSV);   // valid at lanes {row, row+16}
        if (lane < 16) mnormLds[mbase + row] = rnorm;

        // D layout: lane = N (head), VGPR j -> M = j + 8*kh  => contiguous 8 floats per lane
        if (hval < HH) {
            float* dst = &dotLds[hval * SSTR + mbase + (kh << 3)];
            v4f lo = { c[0], c[1], c[2], c[3] };
            v4f hi = { c[4], c[5], c[6], c[7] };
            *(v4f*)(dst)     = lo;
            *(v4f*)(dst + 4) = hi;
        }
    }
    __syncthreads();

    // ---- phase 2: dense normalize + softmax over M per head, from LDS ----
    float kn4[HH], sp4[HH];
#pragma unroll
    for (int h = 0; h < HH; ++h) { kn4[h] = red[48 + h]; sp4[h] = red[52 + h]; }
    float mn[8];
#pragma unroll
    for (int i = 0; i < 8; ++i) mn[i] = mnormLds[tid + i * NTHREADS];

    float* outb = out + (size_t)b * HH * MM;
    for (int h = 0; h < HH; ++h) {
        float vals[8];
        float lmax = -3.402823466e38f;
#pragma unroll
        for (int i = 0; i < 8; ++i) {
            const float d = dotLds[h * SSTR + tid + i * NTHREADS];
            vals[i] = d * sp4[h] * __builtin_amdgcn_rcpf(fmaf(kn4[h], mn[i], EPSV));
            lmax = fmaxf(lmax, vals[i]);
        }
        lmax = wave_max(lmax);
        if (lane == 0) red[wid] = lmax;
        __syncthreads();
        if (tid < NWAVES) {
            float v = wave_max(red[tid]);
            if (tid == 0) red[32] = v;
        }
        __syncthreads();
        const float mx = red[32];

        float lsum = 0.f;
#pragma unroll
        for (int i = 0; i < 8; ++i) { vals[i] = __expf(vals[i] - mx); lsum += vals[i]; }
        lsum = wave_sum(lsum);
        if (lane == 0) red[wid] = lsum;
        __syncthreads();
        if (tid < NWAVES) {
            float v = wave_sum(red[tid]);
            if (tid == 0) red[33] = v;
        }
        __syncthreads();
        const float inv = __builtin_amdgcn_rcpf(red[33]);
#pragma unroll
        for (int i = 0; i < 8; ++i)
            outb[(size_t)h * MM + tid + i * NTHREADS] = vals[i] * inv;
        __syncthreads();   // red[0..33] reused next head (disjoint from red[48..55])
    }
}

extern "C" void kernel_launch(void* const* d_in, const int* in_sizes, int n_in,
                              void* d_out, int out_size, void* d_ws, size_t ws_size,
                              hipStream_t stream) {
    const float* memory    = (const float*)d_in[0];   // [B, M, W] f32
    const float* keys      = (const float*)d_in[1];   // [B, H, W] f32
    const float* strengths = (const float*)d_in[2];   // [B, H]    f32
    float* out = (float*)d_out;                       // [B, H, M] f32

    const size_t shmem = (size_t)(4 * SSTR + MM + 64) * sizeof(float);
    cosine_addr_softmax_kernel<<<BB, NTHREADS, shmem, stream>>>(memory, keys, strengths, out);
}